// GeometryPreservingDiffusionLoss_32057635897817
// MI455X (gfx1250) — compile-verified
//
#include <hip/hip_runtime.h>
#include <math.h>

typedef __attribute__((ext_vector_type(2))) float v2f;
typedef __attribute__((ext_vector_type(8))) float v8f;

#define BN 8
#define NN 4096
#define CN 256
#define DN 1024

// workspace layout (float offsets)
#define CONT_OFF 0          // 2048*6 = 12288
#define DIFF_OFF 12288      // 8
#define SHA_OFF  12296      // 8*30 = 240
#define SHB_OFF  12536      // 8*4  = 32
#define LOC_OFF  12568      // 8*32 = 256
#define SMO_OFF  12824      // 8*32*2 = 512
#define STY_OFF  13336      // 8*3 = 24
// total = 13360 floats (~53 KB)

__device__ __forceinline__ float bsum256(float v, float* red) {
  const int tid = threadIdx.x;
  __syncthreads();
  red[tid] = v;
  __syncthreads();
#pragma unroll
  for (int s = 128; s > 0; s >>= 1) {
    if (tid < s) red[tid] += red[tid + s];
    __syncthreads();
  }
  return red[0];
}

// ---------------- diffusion MSE partials ----------------
__global__ __launch_bounds__(256) void diff_kernel(const float* __restrict__ a,
                                                   const float* __restrict__ b,
                                                   int n, float* __restrict__ out) {
  __shared__ float red[256];
  float acc = 0.f;
  for (int t = blockIdx.x * 256 + threadIdx.x; t < n; t += gridDim.x * 256) {
    float d = a[t] - b[t];
    acc += d * d;
  }
  float s = bsum256(acc, red);
  if (threadIdx.x == 0) out[blockIdx.x] = s;
}

// ---------------- content per-row sums ----------------
// per (b,c) row of length 4096: [sum_o, sum_oo, sum_n, sum_nn, sum_diff2, sum_abs_o]
__global__ __launch_bounds__(256) void content_kernel(const float* __restrict__ o,
                                                      const float* __restrict__ nsy,
                                                      float* __restrict__ out) {
  __shared__ float red[256];
  const int row = blockIdx.x;
  const float* po = o + (size_t)row * NN;
  const float* pn = nsy + (size_t)row * NN;
  float so = 0, soo = 0, sn = 0, snn = 0, sd = 0, sa = 0;
  for (int t = threadIdx.x; t < NN; t += 256) {
    float x = po[t], y = pn[t];
    so += x; soo += x * x; sn += y; snn += y * y;
    float d = y - x; sd += d * d; sa += fabsf(x);
  }
  float r0 = bsum256(so, red), r1 = bsum256(soo, red), r2 = bsum256(sn, red),
        r3 = bsum256(snn, red), r4 = bsum256(sd, red), r5 = bsum256(sa, red);
  if (threadIdx.x == 0) {
    float* q = out + (size_t)row * 6;
    q[0] = r0; q[1] = r1; q[2] = r2; q[3] = r3; q[4] = r4; q[5] = r5;
  }
}

// ---------------- shape phase A: per-batch moments + min/max ----------------
// per tensor 15 values: [sx,sy,sz, sxx,syy,szz, sxy,sxz,syz, mnx,mny,mnz, mxx,mxy,mxz]
__global__ __launch_bounds__(256) void shapeA_kernel(const float* __restrict__ gen,
                                                     const float* __restrict__ orig,
                                                     float* __restrict__ out) {
  __shared__ float red[256];
  const int b = blockIdx.x, tid = threadIdx.x;
  float acc[30];
#pragma unroll
  for (int q = 0; q < 30; ++q) {
    int qt = q % 15;
    acc[q] = (qt < 9) ? 0.f : ((qt < 12) ? 3.4e38f : -3.4e38f);
  }
#pragma unroll
  for (int p = 0; p < 2; ++p) {
    const float* P = (p == 0 ? gen : orig) + (size_t)b * NN * 3;
    const int q0 = p * 15;
    for (int t = tid; t < NN; t += 256) {
      float x = P[3 * t], y = P[3 * t + 1], z = P[3 * t + 2];
      acc[q0 + 0] += x; acc[q0 + 1] += y; acc[q0 + 2] += z;
      acc[q0 + 3] += x * x; acc[q0 + 4] += y * y; acc[q0 + 5] += z * z;
      acc[q0 + 6] += x * y; acc[q0 + 7] += x * z; acc[q0 + 8] += y * z;
      acc[q0 + 9]  = fminf(acc[q0 + 9], x);
      acc[q0 + 10] = fminf(acc[q0 + 10], y);
      acc[q0 + 11] = fminf(acc[q0 + 11], z);
      acc[q0 + 12] = fmaxf(acc[q0 + 12], x);
      acc[q0 + 13] = fmaxf(acc[q0 + 13], y);
      acc[q0 + 14] = fmaxf(acc[q0 + 14], z);
    }
  }
#pragma unroll
  for (int q = 0; q < 30; ++q) {
    const int qt = q % 15;
    __syncthreads();
    red[tid] = acc[q];
    __syncthreads();
#pragma unroll
    for (int s = 128; s > 0; s >>= 1) {
      if (tid < s) {
        float u = red[tid], w = red[tid + s];
        red[tid] = (qt < 9) ? (u + w) : ((qt < 12) ? fminf(u, w) : fmaxf(u, w));
      }
      __syncthreads();
    }
    if (tid == 0) out[b * 30 + q] = red[0];
  }
}

// ---------------- shape phase B: per-batch sum(gd), sum(gd^2) ----------------
__global__ __launch_bounds__(256) void shapeB_kernel(const float* __restrict__ gen,
                                                     const float* __restrict__ orig,
                                                     const float* __restrict__ sha,
                                                     float* __restrict__ out) {
  __shared__ float red[256];
  const int b = blockIdx.x, tid = threadIdx.x;
  const float inv = 1.f / (float)NN;
  const float* gs = sha + b * 30;
  const float* os = gs + 15;
  const float gcx = gs[0] * inv, gcy = gs[1] * inv, gcz = gs[2] * inv;
  const float ocx = os[0] * inv, ocy = os[1] * inv, ocz = os[2] * inv;
  const float* G = gen + (size_t)b * NN * 3;
  const float* O = orig + (size_t)b * NN * 3;
  float sg = 0, sg2 = 0, so = 0, so2 = 0;
  for (int t = tid; t < NN; t += 256) {
    float dx = G[3 * t] - gcx, dy = G[3 * t + 1] - gcy, dz = G[3 * t + 2] - gcz;
    float g = sqrtf(dx * dx + dy * dy + dz * dz);
    sg += g; sg2 += g * g;
    dx = O[3 * t] - ocx; dy = O[3 * t + 1] - ocy; dz = O[3 * t + 2] - ocz;
    float o = sqrtf(dx * dx + dy * dy + dz * dz);
    so += o; so2 += o * o;
  }
  float r0 = bsum256(sg, red), r1 = bsum256(sg2, red),
        r2 = bsum256(so, red), r3 = bsum256(so2, red);
  if (tid == 0) {
    float* q = out + b * 4;
    q[0] = r0; q[1] = r1; q[2] = r2; q[3] = r3;
  }
}

// ---------------- style per-batch dot / |a|^2 / |b|^2 ----------------
__global__ __launch_bounds__(256) void style_kernel(const float* __restrict__ a,
                                                    const float* __restrict__ b,
                                                    float* __restrict__ out) {
  __shared__ float red[256];
  const int bb = blockIdx.x, tid = threadIdx.x;
  const float* pa = a + (size_t)bb * DN;
  const float* pb = b + (size_t)bb * DN;
  float sd = 0, sa2 = 0, sb2 = 0;
  for (int t = tid; t < DN; t += 256) {
    float x = pa[t], y = pb[t];
    sd += x * y; sa2 += x * x; sb2 += y * y;
  }
  float r0 = bsum256(sd, red), r1 = bsum256(sa2, red), r2 = bsum256(sb2, red);
  if (tid == 0) {
    float* q = out + bb * 3;
    q[0] = r0; q[1] = r1; q[2] = r2;
  }
}

// ---------------- WMMA-based kNN kernel (register-direct) ----------------
// Role-swapped WMMA: A holds 16 candidate points j as (x_j, y_j, z_j, |p_j|^2),
// B holds the wave's 16 query points i as (-2x_i, -2y_i, -2z_i, 1), so
//   D[M][N] = |p_j|^2 - 2 p_j.p_i    (j = j0+M, i = i0+N)
// Per the 16x16 f32 C/D layout, lane l owns column N = l&15 (a fixed query
// point) and rows M = 8*(l>>4)+r in its v8f result: the 8 candidate distances
// for lane's own query arrive directly in registers -> zero LDS roundtrip and
// zero barriers in the scan loop. d2 = sq_i(lane reg) + D. The two lane-halves
// scan disjoint j subsets; their top-K lists are merged once via __shfl.
template <int K, bool LOCAL>
__global__ __launch_bounds__(256) void knn_kernel(const float* __restrict__ sel,
                                                  const float* __restrict__ gpts,
                                                  float* __restrict__ out) {
  __shared__ float2 lxy[NN];  // 32 KB
  __shared__ float lz[NN];    // 16 KB
  __shared__ float red[256];  // 1 KB
  const int b = blockIdx.y;
  const int tid = threadIdx.x;
  const float* P = sel + (size_t)b * NN * 3;
  const float* G = gpts + (size_t)b * NN * 3;
  for (int t = tid; t < NN; t += 256) {
    float x = P[3 * t], y = P[3 * t + 1], z = P[3 * t + 2];
    lxy[t] = make_float2(x, y);
    lz[t] = z;
    if (LOCAL) __builtin_prefetch(&G[3 * t], 0, 0);
  }
  __syncthreads();

  const int lane = tid & 31;
  const int wave = tid >> 5;
  const int m = lane & 15;
  const bool hi = lane >= 16;
  const int i0 = blockIdx.x * 128 + wave * 16;
  const int myi = i0 + m;

  // B fragment (fixed): this wave's 16 query points.
  const float2 pq = lxy[myi];
  const float pz = lz[myi];
  const float sqi = pq.x * pq.x + pq.y * pq.y + pz * pz;
  v2f bfrag;
  if (!hi) { bfrag.x = -2.f * pq.x; bfrag.y = -2.f * pq.y; }
  else     { bfrag.x = -2.f * pz;   bfrag.y = 1.f; }

  float bestd[K];
  int bestj[K];
#pragma unroll
  for (int k = 0; k < K; ++k) { bestd[k] = 3.4e38f; bestj[k] = 0; }

  for (int j0 = 0; j0 < NN; j0 += 16) {
    // A fragment: 16 candidate points (rows M), split across lane halves.
    const int ja = j0 + m;
    const float2 qq = lxy[ja];
    v2f afrag;
    if (!hi) {
      afrag.x = qq.x; afrag.y = qq.y;
    } else {
      const float qz = lz[ja];
      afrag.x = qz;
      afrag.y = qq.x * qq.x + qq.y * qq.y + qz * qz;  // |p_j|^2
    }
    v8f c = {0.f, 0.f, 0.f, 0.f, 0.f, 0.f, 0.f, 0.f};
    v8f d = __builtin_amdgcn_wmma_f32_16x16x4_f32(false, afrag, false, bfrag,
                                                  (short)0, c, false, false);
    const int jb = j0 + (hi ? 8 : 0);
#pragma unroll
    for (int r = 0; r < 8; ++r) {
      const int jj = jb + r;
      const float d2 = sqi + d[r];
      if (jj != myi && d2 < bestd[K - 1]) {
        bestd[K - 1] = d2; bestj[K - 1] = jj;
#pragma unroll
        for (int k = K - 1; k > 0; --k) {
          if (bestd[k] < bestd[k - 1]) {
            float td = bestd[k]; bestd[k] = bestd[k - 1]; bestd[k - 1] = td;
            int tj = bestj[k]; bestj[k] = bestj[k - 1]; bestj[k - 1] = tj;
          }
        }
      }
    }
  }

  // Merge the hi-half list into the lo-half list (register shuffles only).
#pragma unroll
  for (int k = 0; k < K; ++k) {
    const float od = __shfl(bestd[k], m + 16, 32);
    const int oj = __shfl(bestj[k], m + 16, 32);
    if (!hi && od < bestd[K - 1]) {
      bestd[K - 1] = od; bestj[K - 1] = oj;
#pragma unroll
      for (int k2 = K - 1; k2 > 0; --k2) {
        if (bestd[k2] < bestd[k2 - 1]) {
          float td = bestd[k2]; bestd[k2] = bestd[k2 - 1]; bestd[k2 - 1] = td;
          int tj = bestj[k2]; bestj[k2] = bestj[k2 - 1]; bestj[k2 - 1] = tj;
        }
      }
    }
  }

  float acc = 0.f, acc2 = 0.f;
  if (!hi) {
    if (LOCAL) {
      const float gix = G[3 * myi], giy = G[3 * myi + 1], giz = G[3 * myi + 2];
      const float oix = pq.x, oiy = pq.y, oiz = pz;
#pragma unroll
      for (int k = 0; k < K; ++k) {
        const int j = bestj[k];
        const float2 oj2 = lxy[j];
        float ex = (G[3 * j] - gix) - (oj2.x - oix);
        float ey = (G[3 * j + 1] - giy) - (oj2.y - oiy);
        float ez = (G[3 * j + 2] - giz) - (lz[j] - oiz);
        acc += ex * ex + ey * ey + ez * ez;
      }
    } else {
      float s = 0.f;
#pragma unroll
      for (int k = 0; k < K; ++k) s += sqrtf(fmaxf(bestd[k], 0.f));
      const float mi = s * (1.f / (float)K);
      acc = mi; acc2 = mi * mi;
    }
  }

  float tot = bsum256(acc, red);
  if (LOCAL) {
    if (tid == 0) out[b * gridDim.x + blockIdx.x] = tot;
  } else {
    float tot2 = bsum256(acc2, red);
    if (tid == 0) {
      float* q = out + (b * gridDim.x + blockIdx.x) * 2;
      q[0] = tot; q[1] = tot2;
    }
  }
}

// ---------------- final combine ----------------
__global__ __launch_bounds__(256) void combine_kernel(const float* __restrict__ ws,
                                                      float* __restrict__ out) {
  __shared__ float red[256];
  __shared__ float sbuf[64];
  const int tid = threadIdx.x;

  float msesum = 0, varo = 0, varn = 0, absum = 0;
  for (int r = tid; r < BN * CN; r += 256) {
    const float* q = ws + CONT_OFF + (size_t)r * 6;
    msesum += q[4];
    absum += q[5];
    varo += (q[1] - q[0] * q[0] * (1.f / NN)) * (1.f / (NN - 1));
    varn += (q[3] - q[2] * q[2] * (1.f / NN)) * (1.f / (NN - 1));
  }
  msesum = bsum256(msesum, red);
  varo = bsum256(varo, red);
  varn = bsum256(varn, red);
  absum = bsum256(absum, red);

  float lsum = ws[LOC_OFF + tid];  // exactly 256 partials
  lsum = bsum256(lsum, red);
  float dsum = (tid < 8) ? ws[DIFF_OFF + tid] : 0.f;
  dsum = bsum256(dsum, red);

  if (tid < 8) {
    const int b = tid;
    // smooth: per-batch ddof=1 std of per-point means
    float sm = 0, sm2 = 0;
    for (int k = 0; k < 32; ++k) {
      sm += ws[SMO_OFF + (b * 32 + k) * 2];
      sm2 += ws[SMO_OFF + (b * 32 + k) * 2 + 1];
    }
    float var = (sm2 - sm * sm * (1.f / NN)) * (1.f / (NN - 1));
    sbuf[b] = sqrtf(fmaxf(var, 0.f));

    // shape pieces
    const float* g = ws + SHA_OFF + b * 30;
    const float* o = g + 15;
    const float inv = 1.f / NN;
    const float ninv1 = 1.f / (NN - 1);
    float gc[3] = {g[0] * inv, g[1] * inv, g[2] * inv};
    float oc[3] = {o[0] * inv, o[1] * inv, o[2] * inv};
    float csq = 0, rsq = 0;
#pragma unroll
    for (int d = 0; d < 3; ++d) {
      float e = gc[d] - oc[d];
      csq += e * e;
      float er = (g[12 + d] - g[9 + d]) - (o[12 + d] - o[9 + d]);
      rsq += er * er;
    }
    float f2 = 0;
    {
      float gd0 = (g[3] - NN * gc[0] * gc[0]) * ninv1, od0 = (o[3] - NN * oc[0] * oc[0]) * ninv1;
      float gd1 = (g[4] - NN * gc[1] * gc[1]) * ninv1, od1 = (o[4] - NN * oc[1] * oc[1]) * ninv1;
      float gd2 = (g[5] - NN * gc[2] * gc[2]) * ninv1, od2 = (o[5] - NN * oc[2] * oc[2]) * ninv1;
      float e0 = gd0 - od0, e1 = gd1 - od1, e2 = gd2 - od2;
      f2 += e0 * e0 + e1 * e1 + e2 * e2;
      float gxy = (g[6] - NN * gc[0] * gc[1]) * ninv1, oxy = (o[6] - NN * oc[0] * oc[1]) * ninv1;
      float gxz = (g[7] - NN * gc[0] * gc[2]) * ninv1, oxz = (o[7] - NN * oc[0] * oc[2]) * ninv1;
      float gyz = (g[8] - NN * gc[1] * gc[2]) * ninv1, oyz = (o[8] - NN * oc[1] * oc[2]) * ninv1;
      float exy = gxy - oxy, exz = gxz - oxz, eyz = gyz - oyz;
      f2 += 2.f * (exy * exy + exz * exz + eyz * eyz);
    }
    sbuf[8 + b] = csq;
    sbuf[16 + b] = rsq;
    sbuf[24 + b] = sqrtf(f2);

    const float* sb = ws + SHB_OFF + b * 4;
    float mg = sb[0] * inv, mo = sb[2] * inv;
    float vg = (sb[1] - sb[0] * sb[0] * inv) * ninv1;
    float vo = (sb[3] - sb[2] * sb[2] * inv) * ninv1;
    float sg = sqrtf(fmaxf(vg, 0.f)), sO = sqrtf(fmaxf(vo, 0.f));
    float em = mg - mo, es = sg - sO;
    sbuf[32 + b] = em * em;
    sbuf[40 + b] = es * es;

    const float* st = ws + STY_OFF + b * 3;
    float na = fmaxf(sqrtf(st[1]), 1e-8f), nb = fmaxf(sqrtf(st[2]), 1e-8f);
    sbuf[48 + b] = fabsf(st[0] / (na * nb) - 0.5f);
  }
  __syncthreads();
  if (tid == 0) {
    float diff_loss = dsum / (float)(BN * NN * 3);
    float mse = msesum / (float)(BN * CN * NN);
    float ov = varo / (float)(BN * CN), nv = varn / (float)(BN * CN);
    float act = fmaxf(1.f - absum / (float)(BN * CN * NN), 0.f) * 0.1f;
    float content = mse + fmaxf(0.1f - ov, 0.f) + fmaxf(0.1f - nv, 0.f) + act;
    float csum = 0, rsum = 0, fsum = 0, dmm = 0, dss = 0, smo = 0, sty = 0;
    for (int b = 0; b < 8; ++b) {
      smo += sbuf[b];
      csum += sbuf[8 + b];
      rsum += sbuf[16 + b];
      fsum += sbuf[24 + b];
      dmm += sbuf[32 + b];
      dss += sbuf[40 + b];
      sty += sbuf[48 + b];
    }
    float shape = csum / 24.f + 0.5f * (rsum / 24.f) + 0.1f * (fsum / 8.f) +
                  0.5f * (dmm / 8.f + dss / 8.f);
    float local = lsum / (float)(BN * NN * 16 * 3);
    float smooth = smo / 8.f;
    float style = sty * 0.125f;
    float total = diff_loss + content + 2.f * shape + local + 0.5f * smooth + 0.1f * style;
    out[0] = diff_loss;
    out[1] = content;
    out[2] = shape;
    out[3] = local;
    out[4] = smooth;
    out[5] = style;
    out[6] = total;
  }
}

extern "C" void kernel_launch(void* const* d_in, const int* in_sizes, int n_in,
                              void* d_out, int out_size, void* d_ws, size_t ws_size,
                              hipStream_t stream) {
  (void)n_in; (void)out_size; (void)ws_size;
  const float* pred = (const float*)d_in[0];
  const float* targ = (const float*)d_in[1];
  const float* gen  = (const float*)d_in[2];
  const float* orig = (const float*)d_in[3];
  const float* c_o  = (const float*)d_in[4];
  const float* c_n  = (const float*)d_in[5];
  const float* s_a  = (const float*)d_in[6];
  const float* s_b  = (const float*)d_in[7];
  float* ws = (float*)d_ws;
  float* out = (float*)d_out;

  diff_kernel<<<8, 256, 0, stream>>>(pred, targ, in_sizes[0], ws + DIFF_OFF);
  content_kernel<<<BN * CN, 256, 0, stream>>>(c_o, c_n, ws + CONT_OFF);
  shapeA_kernel<<<BN, 256, 0, stream>>>(gen, orig, ws + SHA_OFF);
  shapeB_kernel<<<BN, 256, 0, stream>>>(gen, orig, ws + SHA_OFF, ws + SHB_OFF);
  knn_kernel<16, true><<<dim3(32, BN), 256, 0, stream>>>(orig, gen, ws + LOC_OFF);
  knn_kernel<4, false><<<dim3(32, BN), 256, 0, stream>>>(gen, gen, ws + SMO_OFF);
  style_kernel<<<BN, 256, 0, stream>>>(s_a, s_b, ws + STY_OFF);
  combine_kernel<<<1, 256, 0, stream>>>(ws, out);
}